// SpatialSparseTokenHandler_77816217469417
// MI455X (gfx1250) — compile-verified
//
#include <hip/hip_runtime.h>
#include <math.h>

// Shapes fixed by the reference
#define BB 8
#define TT 16
#define CC 64
#define NN 4096          // H*W
#define KKEEP 2048       // N * (1 - SPARSE_RATIO)

typedef __attribute__((ext_vector_type(2))) float v2f;
typedef __attribute__((ext_vector_type(8))) float v8f;

// ---------------------------------------------------------------------------
// Pass 1: s[b][t][n] = sqrt( sum_c x[b][t][c][n]^2 )
// grid: (B*T*4) blocks x 256 threads; each thread handles 4 consecutive n (float4)
// ---------------------------------------------------------------------------
__global__ void k_rownorm(const float* __restrict__ x, float* __restrict__ s) {
    int bt      = blockIdx.x >> 2;            // 0..127 == b*T + t
    int quarter = blockIdx.x & 3;
    int n4      = quarter * 256 + threadIdx.x; // float4 index 0..1023
    const float* p = x + (size_t)bt * CC * NN + (size_t)n4 * 4;

    float ax = 0.f, ay = 0.f, az = 0.f, aw = 0.f;
    #pragma unroll 8
    for (int c = 0; c < CC; ++c) {
        // streaming prefetch a few lines ahead (global_prefetch_b8); OOB prefetch
        // at the tail is speculative and silently dropped by HW.
        __builtin_prefetch(p + (size_t)(c + 8) * NN, 0, 0);
        float4 v = *(const float4*)(p + (size_t)c * NN);
        ax = fmaf(v.x, v.x, ax);
        ay = fmaf(v.y, v.y, ay);
        az = fmaf(v.z, v.z, az);
        aw = fmaf(v.w, v.w, aw);
    }
    float4 r = { sqrtf(ax), sqrtf(ay), sqrtf(az), sqrtf(aw) };
    *(float4*)(s + (size_t)bt * NN + (size_t)n4 * 4) = r;
}

// ---------------------------------------------------------------------------
// Pass 2: scores[b][n] = (1/16) * sum_t s[b][t][n]
// Implemented as ones(16x16) @ S(16x16-tile) via 4 accumulating
// V_WMMA_F32_16X16X4_F32 (K=4 each, K total = 16 = T).
// One wave per 16-column tile. 2048 tiles total.
// A (16x4) = all ones. B (4x16) lane layout (mirror of documented A 16x4):
//   lanes 0-15 : N=lane,     VGPR0=K0, VGPR1=K1
//   lanes 16-31: N=lane-16,  VGPR0=K2, VGPR1=K3
// D row M is identical for all M (ones-A), so lanes 0-15 VGPR0 hold the sums.
// ---------------------------------------------------------------------------
__global__ void k_tsum_wmma(const float* __restrict__ s, float* __restrict__ scores) {
    int wave = threadIdx.x >> 5;
    int lane = threadIdx.x & 31;
    int tile = blockIdx.x * (blockDim.x >> 5) + wave;  // 0..2047
    int b    = tile >> 8;                              // 256 tiles per batch
    int n0   = (tile & 255) << 4;
    int half = lane >> 4;
    int col  = lane & 15;

    const float* sb = s + (size_t)b * TT * NN + n0 + col;

    v8f acc = {};
    v2f a;  a[0] = 1.0f; a[1] = 1.0f;
    #pragma unroll
    for (int j = 0; j < 4; ++j) {
        int k0 = 4 * j + 2 * half;                 // this lane's two K rows
        v2f bv;
        bv[0] = sb[(size_t)(k0)     * NN];
        bv[1] = sb[(size_t)(k0 + 1) * NN];
        acc = __builtin_amdgcn_wmma_f32_16x16x4_f32(
                  false, a, false, bv, (short)0, acc, false, false);
    }
    if (half == 0)
        scores[(size_t)b * NN + n0 + col] = acc[0] * (1.0f / 16.0f);
}

// ---------------------------------------------------------------------------
// Pass 3: exact top-k membership via rank.
// rank(n) = #{i : s[i] > s[n]} + #{i < n : s[i] == s[n]}  (== jax top_k order)
// keep iff rank < KKEEP. One block per (b, 256-token chunk); scores in LDS.
// ---------------------------------------------------------------------------
__global__ void k_select(const float* __restrict__ scores, float* __restrict__ mask) {
    __shared__ float ls[NN];
    int b     = blockIdx.x >> 4;
    int chunk = blockIdx.x & 15;

    for (int i = threadIdx.x; i < NN; i += 256)
        ls[i] = scores[(size_t)b * NN + i];
    __syncthreads();

    int   n  = chunk * 256 + threadIdx.x;
    float sv = ls[n];
    int rank = 0;
    for (int i = 0; i < NN; i += 4) {
        float4 v = *(const float4*)&ls[i];
        rank += (int)((v.x > sv) || ((v.x == sv) && ((i    ) < n)));
        rank += (int)((v.y > sv) || ((v.y == sv) && ((i + 1) < n)));
        rank += (int)((v.z > sv) || ((v.z == sv) && ((i + 2) < n)));
        rank += (int)((v.w > sv) || ((v.w == sv) && ((i + 3) < n)));
    }
    mask[(size_t)b * NN + n] = (rank < KKEEP) ? 1.0f : 0.0f;
}

// ---------------------------------------------------------------------------
// Pass 4: out = x * mask[b][n], float4 streaming. T*C*N = 2^22, N = 2^12.
// ---------------------------------------------------------------------------
__global__ void k_apply(const float* __restrict__ x, const float* __restrict__ mask,
                        float* __restrict__ out) {
    size_t v4 = (size_t)blockIdx.x * blockDim.x + threadIdx.x;
    size_t e  = v4 * 4;
    int b = (int)(e >> 22);
    int n = (int)(e & (NN - 1));
    float4 xv = *(const float4*)(x + e);
    float4 mv = *(const float4*)(mask + (size_t)b * NN + n);
    float4 o  = { xv.x * mv.x, xv.y * mv.y, xv.z * mv.z, xv.w * mv.w };
    *(float4*)(out + e) = o;
}

// ---------------------------------------------------------------------------
extern "C" void kernel_launch(void* const* d_in, const int* in_sizes, int n_in,
                              void* d_out, int out_size, void* d_ws, size_t ws_size,
                              hipStream_t stream) {
    const float* x   = (const float*)d_in[0];
    float*       out = (float*)d_out;

    // workspace layout (floats): s[B*T*N] | scores[B*N] | mask[B*N]
    float* s      = (float*)d_ws;
    float* scores = s + (size_t)BB * TT * NN;     // +2 MB
    float* mask   = scores + (size_t)BB * NN;     // +128 KB

    // Pass 1: per-(b,t,n) channel L2 norm
    k_rownorm<<<BB * TT * 4, 256, 0, stream>>>(x, s);

    // Pass 2: WMMA time-mean -> scores  (2048 tiles, 8 waves/block)
    k_tsum_wmma<<<256, 256, 0, stream>>>(s, scores);

    // Pass 3: exact top-2048 membership -> mask
    k_select<<<BB * 16, 256, 0, stream>>>(scores, mask);

    // Pass 4: masked copy (33.5M elems / 4 per thread)
    const int total4 = (BB * TT * CC * NN) / 4;   // 8,388,608
    k_apply<<<total4 / 256, 256, 0, stream>>>(x, mask, out);
}